// TACTUS_40544491274411
// MI455X (gfx1250) — compile-verified
//
#include <hip/hip_runtime.h>
#include <math.h>

#define BB     2048
#define CC     8
#define DD     768
#define NHALF  (BB*CC)     // 16384 rows per half
#define NTOT   (2*BB)      // 4096
#define TEMP_  0.07f
#define NEGF_  -10.0f

#define MT 4               // 16-row tiles per wave (M)
#define NT 4               // 16-col tiles per wave (N)

typedef __attribute__((ext_vector_type(2))) float v2f;
typedef __attribute__((ext_vector_type(8))) float v8f;

// ---------------------------------------------------------------- helpers
__device__ inline unsigned fkey(float f) {           // monotone float->uint
    unsigned u = __float_as_uint(f);
    return (u & 0x80000000u) ? ~u : (u | 0x80000000u);
}
__device__ inline float finv(unsigned k) {           // inverse of fkey
    unsigned u = (k & 0x80000000u) ? (k & 0x7fffffffu) : ~k;
    return __uint_as_float(u);
}
// A-frag (16x4 f32): lane m=L%16 holds rows, k pair split at lane 16.
// B-frag (4x16) mirrors it when B[k][n] = M[(col0+n)*ld + k] (row-major along k).
__device__ inline v2f load_frag(const float* base, int row0, int ld, int k0, int lane) {
    int m = lane & 15, h = lane >> 4;
    const float* p = base + (size_t)(row0 + m) * ld + k0 + 2 * h;
    v2f r; r.x = p[0]; r.y = p[1];
    return r;
}

// ---------------------------------------------------------------- 1) scores = (z @ query)/t   (one wave per row)
__global__ void k_scores(const float* __restrict__ z, const float* __restrict__ query,
                         const float* __restrict__ attn_temp, float* __restrict__ scores) {
    int wave = (blockIdx.x * blockDim.x + threadIdx.x) >> 5;
    int lane = threadIdx.x & 31;
    if (wave >= 2 * NHALF) return;
    const float* row = z + (size_t)wave * DD;
    float s = 0.f;
    for (int d = lane; d < DD; d += 32) s += row[d] * query[d];
    for (int off = 16; off > 0; off >>= 1) s += __shfl_down(s, off, 32);
    if (lane == 0) scores[wave] = s / attn_temp[0];
}

// ---------------------------------------------------------------- 2) per-half global max of scores
__global__ void k_smax(const float* __restrict__ scores, float* __restrict__ maxes) {
    __shared__ float red[256];
    const float* s = scores + blockIdx.x * NHALF;
    float m = -3.0e38f;
    for (int j = threadIdx.x; j < NHALF; j += 256) m = fmaxf(m, s[j]);
    red[threadIdx.x] = m; __syncthreads();
    for (int off = 128; off > 0; off >>= 1) {
        if (threadIdx.x < off) red[threadIdx.x] = fmaxf(red[threadIdx.x], red[threadIdx.x + off]);
        __syncthreads();
    }
    if (threadIdx.x == 0) maxes[blockIdx.x] = red[0];
}

// ---------------------------------------------------------------- 3) segment softmax + weighted pool
__global__ void k_pool(const float* __restrict__ z, const float* __restrict__ scores,
                       const float* __restrict__ maxes, const int* __restrict__ idx_ori,
                       const int* __restrict__ idx_aug, float* __restrict__ pooled) {
    int g = blockIdx.x, half = blockIdx.y;
    const int* idx = half ? idx_aug : idx_ori;
    int seg = idx[g * CC];
    int rowbase = half * NHALF + g * CC;
    float mx = maxes[half];
    float w[CC]; float denom = 0.f;
    #pragma unroll
    for (int c = 0; c < CC; ++c) { float e = expf(scores[rowbase + c] - mx); w[c] = e; denom += e; }
    float inv = 1.0f / (denom + 1e-8f);
    #pragma unroll
    for (int c = 0; c < CC; ++c) w[c] *= inv;
    float* orow = pooled + (size_t)(half * BB + seg) * DD;
    for (int d = threadIdx.x; d < DD; d += blockDim.x) {
        float acc = 0.f;
        #pragma unroll
        for (int c = 0; c < CC; ++c) acc += w[c] * z[(size_t)(rowbase + c) * DD + d];
        orow[d] = acc;
    }
}

// ---------------------------------------------------------------- 4) f = pooled @ W^T + b
// 4x4 register-blocked f32 WMMA: one wave -> 64x64 output tile,
// 16 independent accumulator chains, 8 loads feed 16 wmma per K-step.
__global__ __launch_bounds__(32)
void k_gemm_wt_bias(const float* __restrict__ A, const float* __restrict__ W,
                    const float* __restrict__ bias, float* __restrict__ out) {
    int lane = threadIdx.x & 31;
    int col0 = blockIdx.x * (16 * NT);
    int row0 = blockIdx.y * (16 * MT);
    v8f acc[MT][NT];
    #pragma unroll
    for (int mi = 0; mi < MT; ++mi)
        #pragma unroll
        for (int ni = 0; ni < NT; ++ni) acc[mi][ni] = (v8f){};
    #pragma unroll 2
    for (int k = 0; k < DD; k += 4) {
        v2f a[MT], b[NT];
        #pragma unroll
        for (int mi = 0; mi < MT; ++mi) a[mi] = load_frag(A, row0 + 16 * mi, DD, k, lane);
        #pragma unroll
        for (int ni = 0; ni < NT; ++ni) b[ni] = load_frag(W, col0 + 16 * ni, DD, k, lane); // B[k][n]=W[col+n][k]
        #pragma unroll
        for (int mi = 0; mi < MT; ++mi)
            #pragma unroll
            for (int ni = 0; ni < NT; ++ni)
                acc[mi][ni] = __builtin_amdgcn_wmma_f32_16x16x4_f32(
                    false, a[mi], false, b[ni], (short)0, acc[mi][ni], false, false);
    }
    int m = lane & 15, h = lane >> 4;
    #pragma unroll
    for (int mi = 0; mi < MT; ++mi)
        #pragma unroll
        for (int ni = 0; ni < NT; ++ni) {
            float bv = bias[col0 + 16 * ni + m];
            #pragma unroll
            for (int i = 0; i < 8; ++i)
                out[(size_t)(row0 + 16 * mi + i + 8 * h) * DD + col0 + 16 * ni + m] = acc[mi][ni][i] + bv;
        }
}

// ---------------------------------------------------------------- 5) row L2-normalize f
__global__ void k_rownorm(float* __restrict__ f) {
    __shared__ float red[256];
    float* row = f + (size_t)blockIdx.x * DD;
    float s = 0.f;
    for (int d = threadIdx.x; d < DD; d += 256) { float v = row[d]; s += v * v; }
    red[threadIdx.x] = s; __syncthreads();
    for (int off = 128; off > 0; off >>= 1) {
        if (threadIdx.x < off) red[threadIdx.x] += red[threadIdx.x + off];
        __syncthreads();
    }
    float inv = 1.0f / sqrtf(red[0]);
    for (int d = threadIdx.x; d < DD; d += 256) row[d] *= inv;
}

// ---------------------------------------------------------------- 6) sim = f @ f^T  (4x4-blocked f32 WMMA, 64x64/wave)
__global__ __launch_bounds__(32)
void k_gemm_sim(const float* __restrict__ f, float* __restrict__ sim) {
    int lane = threadIdx.x & 31;
    int col0 = blockIdx.x * (16 * NT);
    int row0 = blockIdx.y * (16 * MT);
    v8f acc[MT][NT];
    #pragma unroll
    for (int mi = 0; mi < MT; ++mi)
        #pragma unroll
        for (int ni = 0; ni < NT; ++ni) acc[mi][ni] = (v8f){};
    #pragma unroll 2
    for (int k = 0; k < DD; k += 4) {
        v2f a[MT], b[NT];
        #pragma unroll
        for (int mi = 0; mi < MT; ++mi) a[mi] = load_frag(f, row0 + 16 * mi, DD, k, lane);
        #pragma unroll
        for (int ni = 0; ni < NT; ++ni) b[ni] = load_frag(f, col0 + 16 * ni, DD, k, lane);
        #pragma unroll
        for (int mi = 0; mi < MT; ++mi)
            #pragma unroll
            for (int ni = 0; ni < NT; ++ni)
                acc[mi][ni] = __builtin_amdgcn_wmma_f32_16x16x4_f32(
                    false, a[mi], false, b[ni], (short)0, acc[mi][ni], false, false);
    }
    int m = lane & 15, h = lane >> 4;
    #pragma unroll
    for (int mi = 0; mi < MT; ++mi)
        #pragma unroll
        for (int ni = 0; ni < NT; ++ni)
            #pragma unroll
            for (int i = 0; i < 8; ++i)
                sim[(size_t)(row0 + 16 * mi + i + 8 * h) * NTOT + col0 + 16 * ni + m] = acc[mi][ni][i];
}

// ---------------------------------------------------------------- 7) per-row count of safe negatives
__global__ void k_counts(const float* __restrict__ sim, int* __restrict__ cnts) {
    __shared__ unsigned red[256];
    int i = blockIdx.x;
    int pair = (i + BB) % NTOT;
    const float* row = sim + (size_t)i * NTOT;
    unsigned c = 0;
    for (int j = threadIdx.x; j < NTOT; j += 256)
        if (j != i && j != pair && row[j] <= 0.9f) ++c;
    red[threadIdx.x] = c; __syncthreads();
    for (int off = 128; off > 0; off >>= 1) {
        if (threadIdx.x < off) red[threadIdx.x] += red[threadIdx.x + off];
        __syncthreads();
    }
    if (threadIdx.x == 0) cnts[i] = (int)red[0];
}

// ---------------------------------------------------------------- 8) global k = max(1, floor(0.5*mean(counts)))
__global__ void k_calc_k(const int* __restrict__ cnts, int* __restrict__ kval) {
    __shared__ unsigned red[256];
    unsigned s = 0;
    for (int j = threadIdx.x; j < NTOT; j += 256) s += (unsigned)cnts[j];
    red[threadIdx.x] = s; __syncthreads();
    for (int off = 128; off > 0; off >>= 1) {
        if (threadIdx.x < off) red[threadIdx.x] += red[threadIdx.x + off];
        __syncthreads();
    }
    if (threadIdx.x == 0) {
        float mean = (float)red[0] / (float)NTOT;
        int k = (int)floorf(0.5f * mean);
        if (k < 1) k = 1;
        kval[0] = k;
    }
}

// ---------------------------------------------------------------- 9) per-row radix-select top-k + logsumexp term
__global__ void k_select(const float* __restrict__ sim, const int* __restrict__ kptr,
                         float* __restrict__ rterms) {
    __shared__ float    vals[NTOT];      // 16 KB: masked negative values of this row
    __shared__ unsigned hist[256];
    __shared__ float    redf[256];
    __shared__ unsigned redu[256];
    __shared__ float    sh_pos, sh_mneg;
    __shared__ unsigned sh_bin, sh_rem;

    const int i    = blockIdx.x;
    const int tid  = threadIdx.x;
    const int pair = (i + BB) % NTOT;
    const int k    = kptr[0];
    const float invT = 1.0f / TEMP_;
    const float* row = sim + (size_t)i * NTOT;

    // load + mask:  j==i -> sentinel, label/potential-pos -> NEG_FILL
    float lmax = -3.0e38f;
    for (int s = 0; s < NTOT / 256; ++s) {
        int j = tid + s * 256;
        float sv = row[j];
        if (j == pair) sh_pos = sv;                      // raw sim of positive pair
        float v;
        if (j == i)                       v = -3.0e38f;  // excluded entirely
        else if (j == pair || sv > 0.9f)  v = NEGF_;     // label or potential-pos
        else                              v = sv;        // safe negative
        vals[j] = v;
        lmax = fmaxf(lmax, v);
    }
    redf[tid] = lmax; __syncthreads();
    for (int off = 128; off > 0; off >>= 1) {
        if (tid < off) redf[tid] = fmaxf(redf[tid], redf[tid + off]);
        __syncthreads();
    }
    if (tid == 0) sh_mneg = redf[0];
    __syncthreads();

    // 4-pass 8-bit radix select: k-th largest value among vals[]
    unsigned prefHigh = 0, remaining = (unsigned)k;
    for (int d = 3; d >= 0; --d) {
        hist[tid] = 0; __syncthreads();
        int shift = 8 * d;
        for (int s = 0; s < NTOT / 256; ++s) {
            unsigned key = fkey(vals[tid + s * 256]);
            unsigned hi  = (d == 3) ? 0u : (key >> (shift + 8));
            if (hi == prefHigh) atomicAdd(&hist[(key >> shift) & 255u], 1u);
        }
        __syncthreads();
        if (tid == 0) {
            unsigned rem = remaining, bin = 0;
            for (int bn = 255; bn >= 0; --bn) {
                unsigned c = hist[bn];
                if (c >= rem) { bin = (unsigned)bn; break; }
                rem -= c;
            }
            sh_bin = bin; sh_rem = rem;
        }
        __syncthreads();
        prefHigh  = (prefHigh << 8) | sh_bin;
        remaining = sh_rem;
        __syncthreads();
    }
    const unsigned tkey = prefHigh;
    const float    tval = finv(tkey);

    // logsumexp: pos + top-k negatives (ties at threshold counted exactly) + fills
    const float m = fmaxf(sh_pos, sh_mneg) * invT;       // row max of logits
    float    sum_gt = 0.f;
    unsigned cnt_gt = 0;
    for (int s = 0; s < NTOT / 256; ++s) {
        float v = vals[tid + s * 256];
        if (fkey(v) > tkey) { ++cnt_gt; sum_gt += expf(v * invT - m); }
    }
    redf[tid] = sum_gt; redu[tid] = cnt_gt; __syncthreads();
    for (int off = 128; off > 0; off >>= 1) {
        if (tid < off) { redf[tid] += redf[tid + off]; redu[tid] += redu[tid + off]; }
        __syncthreads();
    }
    if (tid == 0) {
        float S = expf(sh_pos * invT - m)
                + redf[0]
                + (float)(k - (int)redu[0]) * expf(tval * invT - m)
                + (float)(NTOT - 1 - k)     * expf(NEGF_ * invT - m);
        rterms[i] = logf(S) + m - sh_pos * invT;         // lse - logits[:,0]
    }
}

// ---------------------------------------------------------------- 10) mean over rows
__global__ void k_meanloss(const float* __restrict__ rterms, float* __restrict__ out) {
    __shared__ float red[256];
    float s = 0.f;
    for (int j = threadIdx.x; j < NTOT; j += 256) s += rterms[j];
    red[threadIdx.x] = s; __syncthreads();
    for (int off = 128; off > 0; off >>= 1) {
        if (threadIdx.x < off) red[threadIdx.x] += red[threadIdx.x + off];
        __syncthreads();
    }
    if (threadIdx.x == 0) out[0] = red[0] / (float)NTOT;
}

// ---------------------------------------------------------------- launch
extern "C" void kernel_launch(void* const* d_in, const int* in_sizes, int n_in,
                              void* d_out, int out_size, void* d_ws, size_t ws_size,
                              hipStream_t stream) {
    (void)in_sizes; (void)n_in; (void)out_size; (void)ws_size;
    const float* z     = (const float*)d_in[0];
    const int*   idx_o = (const int*)  d_in[1];
    const int*   idx_a = (const int*)  d_in[2];
    const float* query = (const float*)d_in[3];
    const float* tattn = (const float*)d_in[4];
    const float* W     = (const float*)d_in[5];
    const float* bias  = (const float*)d_in[6];
    float* out = (float*)d_out;

    // workspace carve-up (floats)
    float* ws      = (float*)d_ws;
    float* scores  = ws;                                   // 2*NHALF
    float* maxes   = scores + 2 * NHALF;                   // 2 (pad 16)
    float* pooled  = maxes + 16;                           // NTOT*DD
    float* fbuf    = pooled + (size_t)NTOT * DD;           // NTOT*DD
    float* simb    = fbuf   + (size_t)NTOT * DD;           // NTOT*NTOT (64 MB)
    int*   cnts    = (int*)(simb + (size_t)NTOT * NTOT);   // NTOT
    int*   kval    = cnts + NTOT;                          // 1 (pad 16)
    float* rterms  = (float*)(kval + 16);                  // NTOT

    // 1) scores for both halves (one wave per row)
    k_scores<<<(2 * NHALF * 32 + 255) / 256, 256, 0, stream>>>(z, query, tattn, scores);
    // 2) per-half global max
    k_smax<<<2, 256, 0, stream>>>(scores, maxes);
    // 3) segment softmax + pooled rows (pooled row index = half*B + idx[group])
    k_pool<<<dim3(BB, 2), 256, 0, stream>>>(z, scores, maxes, idx_o, idx_a, pooled);
    // 4) f = pooled @ W^T + b   (f32 WMMA, 64x64 per wave)
    k_gemm_wt_bias<<<dim3(DD / (16 * NT), NTOT / (16 * MT)), 32, 0, stream>>>(pooled, W, bias, fbuf);
    // 5) row normalize
    k_rownorm<<<NTOT, 256, 0, stream>>>(fbuf);
    // 6) sim = f @ f^T          (f32 WMMA, 64x64 per wave)
    k_gemm_sim<<<dim3(NTOT / (16 * NT), NTOT / (16 * MT)), 32, 0, stream>>>(fbuf, simb);
    // 7) per-row safe-negative counts
    k_counts<<<NTOT, 256, 0, stream>>>(simb, cnts);
    // 8) global k
    k_calc_k<<<1, 256, 0, stream>>>(cnts, kval);
    // 9) per-row top-k threshold + logsumexp terms
    k_select<<<NTOT, 256, 0, stream>>>(simb, kval, rterms);
    // 10) mean loss
    k_meanloss<<<1, 256, 0, stream>>>(rterms, out);
}